// MeanNegDynamicMargin_Loss_69320772158193
// MI455X (gfx1250) — compile-verified
//
#include <hip/hip_runtime.h>

typedef __attribute__((ext_vector_type(2))) float v2f;
typedef __attribute__((ext_vector_type(8))) float v8f;

#define NB       8192
#define DIM      128
#define INV_T    14.285714285714286f   /* 1/0.07 */
#define M_SHIFT  14.285714285714286f   /* fixed LSE shift (upper bound of logits) */
#define CLIPV    0.99999988f           /* 1 - 1e-7 in f32 */
#define MARG     0.15f                 /* MARGIN/LAMBDA = 0.3/2 */
#define CSPLIT   4                     /* column splits */

// ---------------------------------------------------------------------------
// Kernel 1: inverse L2 norms per row; zero stats + output.
// 8 waves/block, one wave per row, lane holds float4 (128 = 32 lanes * 4).
// ---------------------------------------------------------------------------
__global__ void norm_kernel(const float* __restrict__ feat,
                            float* __restrict__ inv_norm,
                            float* __restrict__ stats,
                            float* __restrict__ out) {
    const int lane = threadIdx.x & 31;
    const int wv   = threadIdx.x >> 5;
    const int row  = blockIdx.x * 8 + wv;

    const float4* fp = (const float4*)(feat + row * DIM);
    float4 v = fp[lane];
    float ss = v.x * v.x + v.y * v.y + v.z * v.z + v.w * v.w;
#pragma unroll
    for (int m = 16; m >= 1; m >>= 1) ss += __shfl_xor(ss, m, 32);
    if (lane == 0) inv_norm[row] = 1.0f / fmaxf(sqrtf(ss), 1e-12f);

    // zero the stats region (NB*4 floats) and the scalar output
    int idx = blockIdx.x * blockDim.x + threadIdx.x;
    if (idx < NB * 4) stats[idx] = 0.0f;
    if (idx == 0) *out = 0.0f;
}

// ---------------------------------------------------------------------------
// Kernel 2: fused Gram + contrastive epilogue.
// grid = (128 row-blocks, CSPLIT col-splits), block = 128 threads (4 waves).
// Wave w handles rows m0 + 16w .. +15. 64x128 column tile staged in LDS.
// f32 WMMA 16x16x4: A lane m: (A[m][4k+2h], A[m][4k+2h+1]); B mirrors with
// lane = column. C/D: VGPR r holds row r (lanes 0-15) / row r+8 (lanes 16-31).
// ---------------------------------------------------------------------------
__launch_bounds__(128)
__global__ void loss_kernel(const float* __restrict__ feat,
                            const int*   __restrict__ labels,
                            const float* __restrict__ inv_norm,
                            float* __restrict__ stats) {
    __shared__ float tile[64 * DIM];

    const int tid  = threadIdx.x;
    const int lane = tid & 31;
    const int wv   = tid >> 5;       // wave id 0..3
    const int h    = lane >> 4;      // half of wave
    const int ln   = lane & 15;
    const int m0   = blockIdx.x * 64;
    const int rowA = m0 + wv * 16 + ln;

    // ---- A fragments (reused across all column tiles) ----
    v2f a[32];
    {
        const float sA = inv_norm[rowA];
        const float* ap = feat + rowA * DIM + 2 * h;
#pragma unroll
        for (int kk = 0; kk < 32; ++kk) {
            a[kk].x = ap[kk * 4 + 0] * sA;
            a[kk].y = ap[kk * 4 + 1] * sA;
        }
    }
    // labels of this lane's 8 output rows
    int rlab[8];
#pragma unroll
    for (int r = 0; r < 8; ++r) rlab[r] = labels[m0 + wv * 16 + h * 8 + r];

    float accP[8], accW[8], accT[8], accC[8];
#pragma unroll
    for (int r = 0; r < 8; ++r) { accP[r] = 0.f; accW[r] = 0.f; accT[r] = 0.f; accC[r] = 0.f; }

    const int c0   = blockIdx.y * (NB / CSPLIT);
    const int cEnd = c0 + (NB / CSPLIT);

    for (int n0 = c0; n0 < cEnd; n0 += 64) {
        __syncthreads();
        {   // cooperative load of 64 normalized rows into LDS
            const int r    = tid >> 1;
            const int half = tid & 1;
            const float4* src = (const float4*)(feat + (n0 + r) * DIM + half * 64);
            float4* dst = (float4*)(&tile[r * DIM + half * 64]);
            const float s = inv_norm[n0 + r];
#pragma unroll
            for (int i = 0; i < 16; ++i) {
                float4 t = src[i];
                t.x *= s; t.y *= s; t.z *= s; t.w *= s;
                dst[i] = t;
            }
        }
        __syncthreads();

        for (int sub = 0; sub < 4; ++sub) {
            const int n0s  = n0 + sub * 16;
            const int colg = n0s + ln;               // this lane's global column
            const int clab = labels[colg];
            const float* bp = &tile[(sub * 16 + ln) * DIM + 2 * h];

            v8f acc = {0.f, 0.f, 0.f, 0.f, 0.f, 0.f, 0.f, 0.f};
#pragma unroll
            for (int kk = 0; kk < 32; ++kk) {
                v2f b;
                b.x = bp[kk * 4 + 0];
                b.y = bp[kk * 4 + 1];
                acc = __builtin_amdgcn_wmma_f32_16x16x4_f32(
                    false, a[kk], false, b, (short)0, acc, false, false);
            }
            // fused epilogue on the 8 accumulator rows
#pragma unroll
            for (int r = 0; r < 8; ++r) {
                const int mrow = m0 + wv * 16 + h * 8 + r;
                float s    = acc[r];
                float cosv = -fminf(fmaxf(s, -CLIPV), CLIPV);
                float xneg = cosv * INV_T;
                float em   = __expf(1.0f - cosv);
                float xpos = (cosv - MARG * em) * INV_T;
                bool  labeq = (clab == rlab[r]);
                bool  diag  = (colg == mrow);
                float x = labeq ? xpos : xneg;
                float w = __expf(x - M_SHIFT);
                float pm = (labeq && !diag) ? 1.0f : 0.0f;  // positive mask
                float nm = labeq ? 0.0f : 1.0f;             // negative mask (diag is labeq)
                accP[r] += pm * w;
                accW[r] += nm * w;
                accT[r] += pm * x;
                accC[r] += pm;
            }
        }
    }

    // ---- reduce across the 16 lane-columns of each half, accumulate stats ----
#pragma unroll
    for (int r = 0; r < 8; ++r) {
        float P = accP[r], W = accW[r], T = accT[r], C = accC[r];
#pragma unroll
        for (int m = 8; m >= 1; m >>= 1) {
            P += __shfl_xor(P, m, 32);
            W += __shfl_xor(W, m, 32);
            T += __shfl_xor(T, m, 32);
            C += __shfl_xor(C, m, 32);
        }
        if (ln == 0) {   // lanes 0 and 16 own rows (..+r) and (..+8+r)
            const int mrow = m0 + wv * 16 + h * 8 + r;
            atomicAdd(&stats[mrow * 4 + 0], P);
            atomicAdd(&stats[mrow * 4 + 1], W);
            atomicAdd(&stats[mrow * 4 + 2], T);
            atomicAdd(&stats[mrow * 4 + 3], C);
        }
    }
}

// ---------------------------------------------------------------------------
// Kernel 3: per-row loss -> mean
// ---------------------------------------------------------------------------
__global__ void finalize_kernel(const float* __restrict__ stats,
                                float* __restrict__ out) {
    const int row = blockIdx.x * blockDim.x + threadIdx.x;
    float P = stats[row * 4 + 0];
    float W = stats[row * 4 + 1];
    float T = stats[row * 4 + 2];
    float C = stats[row * 4 + 3];

    float nneg    = (float)(NB - 1) - C;
    float meanNeg = W / (nneg + 1e-8f);
    float denom   = P + meanNeg;
    float sumlp   = T - C * M_SHIFT - C * __logf(denom);
    float cnt     = (C < 1e-6f) ? 1.0f : C;
    float val     = (sumlp / cnt) * (1.0f / (float)NB);

#pragma unroll
    for (int m = 16; m >= 1; m >>= 1) val += __shfl_xor(val, m, 32);
    if ((threadIdx.x & 31) == 0) atomicAdd(out, val);
}

// ---------------------------------------------------------------------------
extern "C" void kernel_launch(void* const* d_in, const int* in_sizes, int n_in,
                              void* d_out, int out_size, void* d_ws, size_t ws_size,
                              hipStream_t stream) {
    const float* feat   = (const float*)d_in[0];
    const int*   labels = (const int*)d_in[1];
    // d_in[2] (sigma) is unused by the reference math.

    float* inv_norm = (float*)d_ws;            // NB floats
    float* stats    = inv_norm + NB;           // NB*4 floats
    float* out      = (float*)d_out;

    norm_kernel<<<NB / 8, 256, 0, stream>>>(feat, inv_norm, stats, out);
    dim3 grid(NB / 64, CSPLIT);
    loss_kernel<<<grid, 128, 0, stream>>>(feat, labels, inv_norm, stats);
    finalize_kernel<<<NB / 256, 256, 0, stream>>>(stats, out);
}